// RoIAlign_withBanks_43241730736892
// MI455X (gfx1250) — compile-verified
//
#include <hip/hip_runtime.h>

// RoIAlign-with-banks for MI455X (gfx1250): separable bilinear pooling executed
// as two chained V_WMMA_F32_16X16X4_F32 contractions per box.
//
// out[n,c,i,j] = Σ_k Px[i,k] Σ_k2 S[c,k,k2] Py[k2,j]
// with k,k2 over the 14 floor/ceil rows/cols and Px/Py the proportion weights.

typedef __attribute__((ext_vector_type(2))) float v2f;
typedef __attribute__((ext_vector_type(8))) float v8f;

typedef __attribute__((address_space(1))) int* gint_ptr;  // global (AS1)
typedef __attribute__((address_space(3))) int* lint_ptr;  // LDS (AS3)

#define NUM_LEVELS 6
#define SSTRIDE 226   // per-channel dword stride of S tile (14*16, padded: gcd(226,64)=2)
#define TSTRIDE 114   // per-channel dword stride of T tile (7*16, padded: gcd(114,64)=2)
#define WAVES 4
#define TILES_PER_WAVE 4   // 4 waves * 4 tiles * 16 channels = 256 channels

#if defined(__has_builtin)
#if __has_builtin(__builtin_amdgcn_global_load_async_to_lds_b32) && \
    __has_builtin(__builtin_amdgcn_s_wait_asynccnt)
#define HAVE_ASYNC_LDS 1
#endif
#endif

__global__ __launch_bounds__(128)
void roialign_wmma_kernel(const float* __restrict__ boxes,
                          const float* __restrict__ f0, const float* __restrict__ f1,
                          const float* __restrict__ f2, const float* __restrict__ f3,
                          const float* __restrict__ f4, const float* __restrict__ f5,
                          float* __restrict__ out, int N)
{
    __shared__ float sS[WAVES][16 * SSTRIDE];  // gathered corner values, [c_local][k][k2(16)]
    __shared__ float sT[WAVES][16 * TSTRIDE];  // intermediate,           [c_local][j][k(16)]
    __shared__ float sPx[14], sPy[14];         // [0..6]=floor weight, [7..13]=ceil weight
    __shared__ int   sRow[14], sCol[14];       // [0..6]=floor coord,  [7..13]=ceil coord

    const int box = blockIdx.x;
    if (box >= N) return;  // uniform per block

    const int tid  = threadIdx.x;
    const int lane = tid & 31;
    const int wave = tid >> 5;
    const int n    = lane & 15;   // N-index / c_local depending on phase
    const int half = lane >> 4;   // K-pair selector within a K=4 block

    // ---- per-box level selection (uniform across block) ----
    const float bx1 = boxes[box * 4 + 0], by1 = boxes[box * 4 + 1];
    const float bx2 = boxes[box * 4 + 2], by2 = boxes[box * 4 + 3];
    const float avg = sqrtf((bx2 - bx1) * (by2 - by1));
    int lvl = (int)floorf((float)(NUM_LEVELS - 1) + log2f(avg));
    lvl = lvl < 0 ? 0 : (lvl > NUM_LEVELS - 1 ? NUM_LEVELS - 1 : lvl);
    const float* fm = f0; int D = 38;
    if      (lvl == 1) { fm = f1; D = 19; }
    else if (lvl == 2) { fm = f2; D = 10; }
    else if (lvl == 3) { fm = f3; D = 5;  }
    else if (lvl == 4) { fm = f4; D = 3;  }
    else if (lvl == 5) { fm = f5; D = 1;  }

    float* S = sS[wave];
    float* T = sT[wave];

    // ---- phase 0: grid coords + proportion weights (threads 0..6 = x, 16..22 = y) ----
    {
        int  i7  = -1;
        bool isx = false;
        if (tid < 7)                    { i7 = tid;      isx = true; }
        else if (tid >= 16 && tid < 23) { i7 = tid - 16; }
        if (i7 >= 0) {
            const float c1 = isx ? bx1 : by1;
            const float c2 = isx ? bx2 : by2;
            float r = c1 + (float)i7 * (c2 - c1) * (1.0f / 6.0f);
            r = fminf(fmaxf(r, 0.0f), 1.0f);
            const float u  = r * (float)(D - 1);
            const float a  = floorf(u);
            const float b  = ceilf(u);
            const float dd = (b - a == 0.0f) ? 1.0f : (b - a);
            const float pa = 1.0f - (u - a) / dd;
            const float pb = 1.0f - pa;
            if (isx) { sRow[i7] = (int)a; sRow[i7 + 7] = (int)b; sPx[i7] = pa; sPx[i7 + 7] = pb; }
            else     { sCol[i7] = (int)a; sCol[i7 + 7] = (int)b; sPy[i7] = pa; sPy[i7 + 7] = pb; }
        }
    }

    // ---- zero the K=14,15 pad entries of S and T once (never written later).
    // Required so garbage (possibly NaN) LDS never feeds a WMMA: 0 x NaN = NaN.
    for (int idx = lane; idx < 16 * 14; idx += 32) {           // S pads: 16 ch x 14 k rows
        const int cl = idx / 14, k = idx - cl * 14;
        S[cl * SSTRIDE + k * 16 + 14] = 0.0f;
        S[cl * SSTRIDE + k * 16 + 15] = 0.0f;
    }
    for (int idx = lane; idx < 16 * 7; idx += 32) {            // T pads: 16 ch x 7 j rows
        const int cl = idx / 7, j = idx - cl * 7;
        T[cl * TSTRIDE + j * 16 + 14] = 0.0f;
        T[cl * TSTRIDE + j * 16 + 15] = 0.0f;
    }
    __syncthreads();

    // ---- build B fragments in VGPRs (sparse weight matrices, no LDS staging) ----
    // B layout (wave32, 4x16 f32): VGPR0 = rows {K0 | K2}, VGPR1 = rows {K1 | K3},
    // N = lane&15 striped across each half (mirrors the documented C/D row striping).
    auto wsel = [&](const float* p, int k) -> float {
        if (n >= 7) return 0.0f;
        if (k == n)     return p[n];      // floor weight for grid index n
        if (k == n + 7) return p[n + 7];  // ceil weight for grid index n
        return 0.0f;
    };
    v2f bY[4], bX[4];
#pragma unroll
    for (int kb4 = 0; kb4 < 4; ++kb4) {
        const int k0 = kb4 * 4 + 2 * half;
        bY[kb4].x = wsel(sPy, k0); bY[kb4].y = wsel(sPy, k0 + 1);
        bX[kb4].x = wsel(sPx, k0); bX[kb4].y = wsel(sPx, k0 + 1);
    }

    const size_t DD    = (size_t)D * (size_t)D;
    float*       obase = out + (size_t)box * 256 * 49;

    for (int t = 0; t < TILES_PER_WAVE; ++t) {
        const int c0 = (wave * TILES_PER_WAVE + t) * 16;

        // ---- gather S[c_local][k][k2<14] from the selected fmap (L2-resident) ----
        // CDNA5 path: scattered per-lane global->LDS async copies (ASYNCcnt).
        for (int idx = lane; idx < 16 * 196; idx += 32) {
            const int cl  = idx / 196;
            const int rem = idx - cl * 196;
            const int k   = rem / 14;
            const int k2  = rem - k * 14;
            float* gp = const_cast<float*>(
                fm + (size_t)(c0 + cl) * DD + (size_t)(sRow[k] * D + sCol[k2]));
            float* lp = S + cl * SSTRIDE + k * 16 + k2;
#if HAVE_ASYNC_LDS
            __builtin_amdgcn_global_load_async_to_lds_b32(
                (gint_ptr)gp, (lint_ptr)lp, 0, 0);
#else
            *lp = *gp;
#endif
        }
#if HAVE_ASYNC_LDS
        __builtin_amdgcn_s_wait_asynccnt(0);
#endif
        __syncthreads();

        // ---- contraction 1: T[c,k,j] = sum_k2 S[c,k,k2] * Py[k2,j] ----
        // A frag: lane (M = n) loads {K=2*half, 2*half+1} of each K-block via b64.
        for (int k = 0; k < 14; ++k) {
            v8f acc = {0.f, 0.f, 0.f, 0.f, 0.f, 0.f, 0.f, 0.f};
            const float* ap = S + n * SSTRIDE + k * 16 + 2 * half;
#pragma unroll
            for (int kb4 = 0; kb4 < 4; ++kb4) {
                v2f a = *(const v2f*)(ap + kb4 * 4);
                acc = __builtin_amdgcn_wmma_f32_16x16x4_f32(
                    false, a, false, bY[kb4], (short)0, acc, false, false);
            }
            if (n < 7) {  // D-matrix: lane holds column N=n for rows M=r+8*half
#pragma unroll
                for (int r = 0; r < 8; ++r)
                    T[(r + 8 * half) * TSTRIDE + n * 16 + k] = acc[r];
            }
        }
        __syncthreads();

        // ---- contraction 2: out[c,i,j] = sum_k T[c,k,j] * Px[i,k] ----
        for (int j = 0; j < 7; ++j) {
            v8f acc = {0.f, 0.f, 0.f, 0.f, 0.f, 0.f, 0.f, 0.f};
            const float* ap = T + n * TSTRIDE + j * 16 + 2 * half;
#pragma unroll
            for (int kb4 = 0; kb4 < 4; ++kb4) {
                v2f a = *(const v2f*)(ap + kb4 * 4);
                acc = __builtin_amdgcn_wmma_f32_16x16x4_f32(
                    false, a, false, bX[kb4], (short)0, acc, false, false);
            }
            if (n < 7) {  // N = i = n; rows M = channel r+8*half
#pragma unroll
                for (int r = 0; r < 8; ++r) {
                    const int c = c0 + r + 8 * half;
                    obase[(size_t)c * 49 + n * 7 + j] = acc[r];
                }
            }
        }
        __syncthreads();
    }
}

extern "C" void kernel_launch(void* const* d_in, const int* in_sizes, int n_in,
                              void* d_out, int out_size, void* d_ws, size_t ws_size,
                              hipStream_t stream) {
    (void)n_in; (void)out_size; (void)d_ws; (void)ws_size;
    const float* boxes = (const float*)d_in[0];
    const float* f0 = (const float*)d_in[1];
    const float* f1 = (const float*)d_in[2];
    const float* f2 = (const float*)d_in[3];
    const float* f3 = (const float*)d_in[4];
    const float* f4 = (const float*)d_in[5];
    const float* f5 = (const float*)d_in[6];
    float* out = (float*)d_out;
    const int N = in_sizes[0] / 4;  // boxes are [N,4]
    roialign_wmma_kernel<<<dim3(N), dim3(128), 0, stream>>>(
        boxes, f0, f1, f2, f3, f4, f5, out, N);
}